// CausalHAttention1D_86122684219921
// MI455X (gfx1250) — compile-verified
//
#include <hip/hip_runtime.h>
#include <math.h>

// ---------------------------------------------------------------------------
// Causal H-attention for MI455X (gfx1250, wave32).
// b*h = 32, n = 8192, d = 64, BLOCK = 16  ->  one wave per 16-row block,
// 16x16x64 GEMMs done with V_WMMA_F32_16X16X4_F32 (f32-exact, 16 WMMA each).
// Memory-bound (~268MB @ 23.3TB/s ~= 11.5us floor); compute trivial.
// Workspace: q/k-mean + v-sum pyramids for levels 3..8  (~49.5 MB f32).
// v3: stage-wise batched ds_swizzle butterflies (8 rows reduced in parallel,
//     one dscnt drain per stage instead of per row), batched v_exp_f32.
// ---------------------------------------------------------------------------

typedef __attribute__((ext_vector_type(2))) float v2f;
typedef __attribute__((ext_vector_type(8))) float v8f;

#define N_SEQ   8192
#define DHEAD   64
#define BHDIM   32
#define NBLK    512                 // N_SEQ / 16 blocks per (b,h)
#define QSTRIDE 68                  // LDS row stride (conflict-free A/B b64 loads)
#define VSTRIDE 72                  // LDS row stride for v (conflict-free B b32 loads)
#define SSTRIDE 20                  // LDS row stride for exp(S) staging
#define WAVE_FLTS (16*QSTRIDE + 16*QSTRIDE + 16*VSTRIDE + 16*QSTRIDE + 16*SSTRIDE + 16)

__device__ __forceinline__ void wave_waitds() {
  // CDNA5 split counters: drain our LDS stores so other lanes' data is visible.
  asm volatile("s_wait_dscnt 0x0" ::: "memory");
}
__device__ __forceinline__ void wait_async() {
  // Drain GLOBAL_LOAD_ASYNC_TO_LDS transfers (ASYNCcnt).
  asm volatile("s_wait_asynccnt 0x0" ::: "memory");
}

// 16-lane XOR butterfly step via DS_SWIZZLE_B32 (group-of-32 mode, and=0x1F).
template <int MASK>
__device__ __forceinline__ float xor_swz(float x) {
  int r = __builtin_amdgcn_ds_swizzle(__float_as_int(x), (MASK << 10) | 0x1F);
  return __int_as_float(r);
}
// Batched butterfly stages over 8 independent rows: 8 swizzles issue
// back-to-back, then 8 consumers -> one LDS latency per stage, not per row.
template <int MASK>
__device__ __forceinline__ void bmax8(float* x) {
  float t[8];
#pragma unroll
  for (int g = 0; g < 8; ++g) t[g] = xor_swz<MASK>(x[g]);
#pragma unroll
  for (int g = 0; g < 8; ++g) x[g] = fmaxf(x[g], t[g]);
}
template <int MASK>
__device__ __forceinline__ void bsum8(float* x) {
  float t[8];
#pragma unroll
  for (int g = 0; g < 8; ++g) t[g] = xor_swz<MASK>(x[g]);
#pragma unroll
  for (int g = 0; g < 8; ++g) x[g] += t[g];
}

// Async-copy a contiguous 16x64 f32 tile straight into padded LDS.
// (flat pointers to __shared__ keep the LDS byte offset in the low 32 bits)
__device__ __forceinline__ void async_tile16x64(const float* __restrict__ src,
                                                float* dst, int stride, int lane) {
  const int r0  = lane >> 4;          // 2 rows / instruction across the wave
  const int col = (lane * 4) & 63;
#pragma unroll
  for (int it = 0; it < 8; ++it) {
    int row = 2 * it + r0;
    unsigned long long g = (unsigned long long)(size_t)(src + row * 64 + col);
    unsigned int       l = (unsigned int)(size_t)(dst + row * stride + col);
    asm volatile("global_load_async_to_lds_b128 %0, %1, off"
                 :: "v"(l), "v"(g) : "memory");
  }
}

// Build a 16x64 coarse tile by reducing F consecutive fine rows per coarse row.
__device__ __forceinline__ void loadAggr(const float* __restrict__ src, float* dst,
                                         int stride, int F, float mul, int lane) {
  for (int mr = 0; mr < 16; ++mr) {
    float s0 = 0.f, s1 = 0.f;
    const float* p = src + (size_t)mr * F * 64;
    for (int f = 0; f < F; ++f) { s0 += p[f*64 + lane]; s1 += p[f*64 + 32 + lane]; }
    dst[mr*stride + lane]      = s0 * mul;
    dst[mr*stride + 32 + lane] = s1 * mul;
  }
}

// One 16x16 (d=64) attention block:  S = scale * (qb @ kb^T)  (f32 WMMA),
// optional causal mask, rowmax-sub + exp, rowsum -> acb,  y = exp(S) @ vb -> ycb.
__device__ __forceinline__ void attn16(const float* qb, const float* kb, const float* vb,
                                       float* sbuf, float* acb, float* ycb,
                                       bool causal, int lane) {
  const int h  = lane >> 4;        // lane half selects K/M sub-pair per WMMA layout
  const int m  = lane & 15;
  const int c0 = 2 * h;

  v8f S = {0.f,0.f,0.f,0.f,0.f,0.f,0.f,0.f};
#pragma unroll
  for (int kk = 0; kk < 16; ++kk) {
    // A (16x4): lane -> row m, K = 4kk + 2h + {0,1};  B (4x16): identical addressing
    v2f a = *(const v2f*)(qb + m*QSTRIDE + 4*kk + c0);
    v2f b = *(const v2f*)(kb + m*QSTRIDE + 4*kk + c0);
    S = __builtin_amdgcn_wmma_f32_16x16x4_f32(false, a, false, b, (short)0, S, false, false);
  }

  // scale (exact pow2) + mask; C-layout: row r = g + 8h, col n = m
  float sv[8], red[8];
#pragma unroll
  for (int g = 0; g < 8; ++g) {
    float s = S[g] * 0.125f;               // d^-0.5
    if (causal && (m > g + 8*h)) s = -3.402823466e38f;
    sv[g] = s; red[g] = s;
  }
  // batched rowmax butterflies (8 rows in flight per stage)
  bmax8<1>(red); bmax8<2>(red); bmax8<4>(red); bmax8<8>(red);
  float e[8], a8[8];
#pragma unroll
  for (int g = 0; g < 8; ++g) { e[g] = __expf(sv[g] - red[g]); a8[g] = e[g]; }
  // batched rowsum butterflies
  bsum8<1>(a8); bsum8<2>(a8); bsum8<4>(a8); bsum8<8>(a8);
#pragma unroll
  for (int g = 0; g < 8; ++g) sbuf[(g + 8*h)*SSTRIDE + m] = e[g];
  if (m == 0) {                            // lanes 0 & 16 publish row sums
#pragma unroll
    for (int g = 0; g < 8; ++g) acb[g + 8*h] = a8[g];
  }
  wave_waitds();

  // y = exp(S) @ v : four 16-wide N tiles, K = 16 in chunks of 4
#pragma unroll
  for (int t = 0; t < 4; ++t) {
    v8f y = {0.f,0.f,0.f,0.f,0.f,0.f,0.f,0.f};
#pragma unroll
    for (int kk = 0; kk < 4; ++kk) {
      v2f a = *(const v2f*)(sbuf + m*SSTRIDE + 4*kk + c0);
      int rb = 4*kk + c0;
      v2f b;
      b.x = vb[(rb + 0)*VSTRIDE + 16*t + m];
      b.y = vb[(rb + 1)*VSTRIDE + 16*t + m];
      y = __builtin_amdgcn_wmma_f32_16x16x4_f32(false, a, false, b, (short)0, y, false, false);
    }
#pragma unroll
    for (int g = 0; g < 8; ++g) ycb[(g + 8*h)*QSTRIDE + 16*t + m] = y[g];
  }
  wave_waitds();
}

// Accumulate a level-j block result into the per-wave fragments.
// Fine row r contributes iff low j bits of p = 16*i + r are all ones;
// source coarse-local row lr = (p >> j) & 15.
__device__ __forceinline__ void accum(int j, int i, int lane, v8f* yacc, float* aacc,
                                      const float* ycb, const float* acb) {
  const int h = lane >> 4, m = lane & 15;
  const int msk = (1 << j) - 1;
#pragma unroll
  for (int g = 0; g < 8; ++g) {
    int p = 16*i + g + 8*h;
    if ((p & msk) == msk) {
      int lr = (p >> j) & 15;
      aacc[g] += acb[lr];
#pragma unroll
      for (int t = 0; t < 4; ++t) yacc[t][g] += ycb[lr*QSTRIDE + 16*t + m];
    }
  }
}

// ---------------------------------------------------------------------------
// Pyramid builder: q,k -> mean of `factor` rows ; v -> sum.  One wave per row.
// ---------------------------------------------------------------------------
__global__ __launch_bounds__(256) void downsample_kernel(
    const float* __restrict__ sq, const float* __restrict__ sk, const float* __restrict__ sv,
    float* __restrict__ dq, float* __restrict__ dk, float* __restrict__ dv,
    int dstRows, int factor) {
  int wid  = blockIdx.x * 8 + (threadIdx.x >> 5);
  int lane = threadIdx.x & 31;
  int bh = wid / dstRows;
  int dr = wid - bh * dstRows;
  size_t srcBase = ((size_t)bh * dstRows + dr) * (size_t)factor * 64;
  size_t dstBase = ((size_t)bh * dstRows + dr) * 64;
  float inv = 1.0f / (float)factor;
  float q0=0,q1=0,k0=0,k1=0,v0=0,v1=0;
  for (int f = 0; f < factor; ++f) {
    size_t o = srcBase + (size_t)f * 64;
    q0 += sq[o+lane]; q1 += sq[o+32+lane];
    k0 += sk[o+lane]; k1 += sk[o+32+lane];
    v0 += sv[o+lane]; v1 += sv[o+32+lane];
  }
  dq[dstBase+lane]=q0*inv; dq[dstBase+32+lane]=q1*inv;
  dk[dstBase+lane]=k0*inv; dk[dstBase+32+lane]=k1*inv;
  dv[dstBase+lane]=v0;     dv[dstBase+32+lane]=v1;
}

// ---------------------------------------------------------------------------
// Fused H-attention: one wave per fine 16-row block. 2 waves / workgroup.
// ---------------------------------------------------------------------------
__global__ __launch_bounds__(64) void hattn_kernel(
    const float* __restrict__ q, const float* __restrict__ k, const float* __restrict__ v,
    float* __restrict__ out, const float* __restrict__ pyr) {
  __shared__ __align__(16) float lds[2 * WAVE_FLTS];
  const int lane = threadIdx.x & 31;
  const int wsl  = threadIdx.x >> 5;
  float* qb   = lds + wsl * WAVE_FLTS;
  float* kb   = qb  + 16*QSTRIDE;
  float* vb   = kb  + 16*QSTRIDE;
  float* ycb  = vb  + 16*VSTRIDE;
  float* sbuf = ycb + 16*QSTRIDE;
  float* acb  = sbuf + 16*SSTRIDE;

  const int wid = blockIdx.x * 2 + wsl;
  const int bh  = wid >> 9;                 // 512 blocks per (b,h)
  const int i   = wid & (NBLK - 1);
  const float* qf = q + (size_t)bh * N_SEQ * DHEAD;
  const float* kf = k + (size_t)bh * N_SEQ * DHEAD;
  const float* vf = v + (size_t)bh * N_SEQ * DHEAD;

  v8f yacc[4];
  float aacc[8];
#pragma unroll
  for (int t = 0; t < 4; ++t) yacc[t] = (v8f){0.f,0.f,0.f,0.f,0.f,0.f,0.f,0.f};
#pragma unroll
  for (int g = 0; g < 8; ++g) aacc[g] = 0.f;

  // ---- level 0: causal diagonal block (async direct-to-LDS stage) ----
  async_tile16x64(qf + (size_t)i*16*64, qb, QSTRIDE, lane);
  async_tile16x64(kf + (size_t)i*16*64, kb, QSTRIDE, lane);
  async_tile16x64(vf + (size_t)i*16*64, vb, VSTRIDE, lane);
  wait_async();
  attn16(qb, kb, vb, sbuf, acb, ycb, true, lane);
  accum(0, i, lane, yacc, aacc, ycb, acb);

  // ---- levels 1..2: aggregate directly from fine data (L2-resident) ----
#pragma unroll
  for (int j = 1; j <= 2; ++j) {
    int cb = i >> j;
    if (cb & 1) {                            // odd coarse block attends left sibling
      int F = 1 << j;
      loadAggr(qf + ((size_t)(16*cb)     << j) * 64, qb, QSTRIDE, F, 1.f/(float)F, lane);
      loadAggr(kf + ((size_t)(16*(cb-1)) << j) * 64, kb, QSTRIDE, F, 1.f/(float)F, lane);
      loadAggr(vf + ((size_t)(16*(cb-1)) << j) * 64, vb, VSTRIDE, F, 1.f,          lane);
      wave_waitds();
      attn16(qb, kb, vb, sbuf, acb, ycb, false, lane);
      accum(j, i, lane, yacc, aacc, ycb, acb);
    }
  }

  // ---- levels 3..8: async-copy precomputed pyramid tiles ----
  size_t off = 0;
#pragma unroll
  for (int j = 3; j <= 8; ++j) {
    size_t per = (size_t)BHDIM * (size_t)(N_SEQ >> j) * DHEAD;
    size_t qo = off, ko = off + per, vo = off + 2*per;
    off += 3 * per;
    int cb = i >> j;
    if (cb & 1) {
      size_t rbase = (size_t)bh * (size_t)(N_SEQ >> j) * DHEAD;
      async_tile16x64(pyr + qo + rbase + (size_t)(16*cb)     * 64, qb, QSTRIDE, lane);
      async_tile16x64(pyr + ko + rbase + (size_t)(16*(cb-1)) * 64, kb, QSTRIDE, lane);
      async_tile16x64(pyr + vo + rbase + (size_t)(16*(cb-1)) * 64, vb, VSTRIDE, lane);
      wait_async();
      attn16(qb, kb, vb, sbuf, acb, ycb, false, lane);
      accum(j, i, lane, yacc, aacc, ycb, acb);
    }
  }

  // ---- normalize + store (C-layout: row g+8h, col 16t+m) ----
  const int h = lane >> 4, m = lane & 15;
  float* ob = out + ((size_t)bh * N_SEQ + (size_t)i * 16) * 64;
#pragma unroll
  for (int g = 0; g < 8; ++g) {
    int r = g + 8*h;
    float inv = 1.f / (aacc[g] + 1e-8f);
#pragma unroll
    for (int t = 0; t < 4; ++t) ob[(size_t)r*64 + 16*t + m] = yacc[t][g] * inv;
  }
}

// ---------------------------------------------------------------------------
extern "C" void kernel_launch(void* const* d_in, const int* in_sizes, int n_in,
                              void* d_out, int out_size, void* d_ws, size_t ws_size,
                              hipStream_t stream) {
  (void)in_sizes; (void)n_in; (void)out_size; (void)ws_size;
  const float* q = (const float*)d_in[0];
  const float* k = (const float*)d_in[1];
  const float* v = (const float*)d_in[2];
  float* out = (float*)d_out;
  float* ws  = (float*)d_ws;   // needs ~49.6 MB: q/k/v pyramids, levels 3..8

  size_t qo[9], ko[9], vo[9], off = 0;
  for (int j = 3; j <= 8; ++j) {
    size_t per = (size_t)32 * (size_t)(8192 >> j) * 64;
    qo[j] = off; ko[j] = off + per; vo[j] = off + 2*per; off += 3*per;
  }

  { // level 3 straight from fine data (factor 8)
    int dstRows = 8192 >> 3;
    downsample_kernel<<<dim3((32*dstRows)/8), dim3(256), 0, stream>>>(
        q, k, v, ws+qo[3], ws+ko[3], ws+vo[3], dstRows, 8);
  }
  for (int j = 4; j <= 8; ++j) {  // halve level by level
    int dstRows = 8192 >> j;
    downsample_kernel<<<dim3((32*dstRows)/8), dim3(256), 0, stream>>>(
        ws+qo[j-1], ws+ko[j-1], ws+vo[j-1], ws+qo[j], ws+ko[j], ws+vo[j], dstRows, 2);
  }

  hattn_kernel<<<dim3(32*512/2), dim3(64), 0, stream>>>(q, k, v, out, ws);
}